// MEOW_37512244363667
// MI455X (gfx1250) — compile-verified
//
#include <hip/hip_runtime.h>

typedef __attribute__((ext_vector_type(2))) float v2f;
typedef __attribute__((ext_vector_type(8))) float v8f;
typedef __attribute__((ext_vector_type(4))) unsigned int u32x4;
typedef __attribute__((ext_vector_type(8))) unsigned int u32x8;

#define DIN    128
#define HDIM   64
#define XPITCH 132   // LDS pitch (floats) for X tile rows: 128 + 4 pad via TDM pad fields
#define MPITCH 68    // LDS pitch for mean rows in finalize

__device__ __forceinline__ float elu1(float x) {
    return x > 0.0f ? x : (__expf(x) - 1.0f);
}

// ---------------- zero workspace (sums + counts) ----------------
__global__ void k_zero(float* __restrict__ p, size_t n) {
    size_t i = (size_t)blockIdx.x * blockDim.x + threadIdx.x;
    size_t stride = (size_t)gridDim.x * blockDim.x;
    for (; i < n; i += stride) p[i] = 0.0f;
}

// ---------------- H = elu(X @ W + b) ----------------
// X: [M,128] rm, W: [128,64] rm, b: [64], out: [M,64]
// 128 threads = 4 waves; block covers 64 rows (16-row slab per wave).
// X tile arrives via the Tensor Data Mover (padded pitch 132 for bank spread,
// zero-fill beyond tensor_dim1 handles the ragged last block).
__global__ __launch_bounds__(128) void k_gemm_elu(
    const float* __restrict__ X, const float* __restrict__ W,
    const float* __restrict__ b, float* __restrict__ out, int M)
{
    extern __shared__ float smem[];
    float* sW = smem;                         // (DIN/2)*HDIM float2, K-pair interleaved
    float* sX = smem + (DIN / 2) * HDIM * 2;  // 64 x XPITCH floats
    const int tid   = threadIdx.x;
    const int wave  = tid >> 5;
    const int lane  = tid & 31;
    const int nlane = lane & 15;
    const int row0  = blockIdx.x * 64;

    if (wave == 0) {
        // --- Tensor DMA descriptor (D#): 2D tile [64 x 128] of f32 ---
        unsigned long long ga = (unsigned long long)(uintptr_t)X +
                                (unsigned long long)row0 * (DIN * 4);
        unsigned int ldsa = (unsigned int)(uintptr_t)sX;   // low 32 bits = LDS byte offset
        unsigned int rows = (unsigned int)(M - row0);      // OOB rows read as zero
        u32x4 g0;
        g0[0] = 1u;                                        // count = 1 (valid descriptor)
        g0[1] = ldsa;                                      // lds_addr
        g0[2] = (unsigned int)ga;                          // global_addr[31:0]
        g0[3] = ((unsigned int)(ga >> 32) & 0x1FFFFFFu)    // global_addr[56:32]
              | 0x80000000u;                               // type = 2 ("image")
        u32x8 g1;
        g1[0] = (2u << 16)      // data_size = 4B
              | (1u << 20)      // pad_enable
              | (6u << 22)      // pad_interval: every 128 DWORDs (one row)
              | (3u << 25);     // pad_amount: 4 DWORDs  -> LDS pitch 132 floats
        g1[1] = ((unsigned)DIN & 0xFFFFu) << 16;                        // tensor_dim0[15:0]
        g1[2] = (((unsigned)DIN >> 16) & 0xFFFFu) | ((rows & 0xFFFFu) << 16); // dim0 hi | dim1 lo
        g1[3] = ((rows >> 16) & 0xFFFFu) | ((unsigned)DIN << 16);       // dim1 hi | tile_dim0=128
        g1[4] = 64u;             // tile_dim1 = 64 rows, tile_dim2 = 0
        g1[5] = (unsigned)DIN;   // tensor_dim0_stride = 128
        g1[6] = 0u;
        g1[7] = 0u;
        asm volatile("tensor_load_to_lds %0, %1" :: "s"(g0), "s"(g1) : "memory");
        __builtin_amdgcn_s_wait_tensorcnt(0);
    }

    // stage W pair-interleaved while the TDM streams X: sW[p][n] = {W[2p][n], W[2p+1][n]}
    for (int i = tid; i < (DIN / 2) * HDIM; i += 128) {
        int p = i >> 6, n = i & 63;
        v2f w;
        w.x = W[(2 * p)     * HDIM + n];
        w.y = W[(2 * p + 1) * HDIM + n];
        ((v2f*)sW)[p * HDIM + n] = w;
    }
    __syncthreads();

    // A frag: lanes 0-15 carry K={k0,k0+1}, lanes 16-31 K={k0+2,k0+3} (ISA 16x4 f32 layout)
    const int    kHalf  = (lane < 16) ? 0 : 2;
    const float* xrow   = sX + (wave * 16 + nlane) * XPITCH;
    const v2f*   wpairs = (const v2f*)sW;

    v8f acc[4] = {};
    for (int k0 = 0; k0 < DIN; k0 += 4) {
        v2f a = *(const v2f*)(xrow + k0 + kHalf);                 // single ds_load_b64
        const v2f* wp = wpairs + (((k0 + kHalf) >> 1) * HDIM + nlane);
#pragma unroll
        for (int t = 0; t < 4; ++t) {
            acc[t] = __builtin_amdgcn_wmma_f32_16x16x4_f32(
                false, a, false, wp[t * 16], (short)0, acc[t], false, false);
        }
    }

    // C layout: VGPR r -> M = r (lanes 0-15) / M = 8+r (lanes 16-31), N = lane&15
    const int rhalf = (lane < 16) ? 0 : 8;
#pragma unroll
    for (int t = 0; t < 4; ++t) {
        int col = t * 16 + nlane;
        float bias = b[col];
#pragma unroll
        for (int r = 0; r < 8; ++r) {
            int row = row0 + wave * 16 + rhalf + r;
            if (row < M)
                out[(size_t)row * HDIM + col] = elu1(acc[t][r] + bias);
        }
    }
}

// ---------------- edge scatter: sum[src] += h_nei[dst]; cnt[src] += 1 ----------------
// One wave per edge: src/dst loads are wave-uniform; each lane atomics a float2 (L2-resident).
__global__ __launch_bounds__(256) void k_edge_agg(
    const float* __restrict__ hnei, const int* __restrict__ src,
    const int* __restrict__ dst, float* __restrict__ sum,
    float* __restrict__ cnt, int E)
{
    long long t = (long long)blockIdx.x * blockDim.x + threadIdx.x;
    int e = (int)(t >> 5);
    if (e >= E) return;
    int lane = (int)(t & 31);
    int s = src[e];
    int d = dst[e];
    float2 hv = ((const float2*)(hnei + (size_t)d * HDIM))[lane];
    float* sp = sum + (size_t)s * HDIM + lane * 2;
    atomicAdd(sp,     hv.x);
    atomicAdd(sp + 1, hv.y);
    if (lane == 0) atomicAdd(cnt + s, 1.0f);
}

// ---------------- mean@A0 / mean@A1 + fused ELU combine ----------------
__global__ __launch_bounds__(128) void k_finalize(
    const float* __restrict__ sum1, const float* __restrict__ cnt1,
    const float* __restrict__ sum2, const float* __restrict__ cnt2,
    const float* __restrict__ A0,   const float* __restrict__ A1,
    const float* __restrict__ h_tar, const float* __restrict__ h_mask,
    float* __restrict__ out, int N)
{
    extern __shared__ float smem[];
    float* sA0 = smem;                          // (HDIM/2)*HDIM float2, K-pair interleaved
    float* sA1 = smem + (HDIM / 2) * HDIM * 2;
    float* sM1 = sA1  + (HDIM / 2) * HDIM * 2;  // 64 x MPITCH
    float* sM2 = sM1  + 64 * MPITCH;
    const int tid   = threadIdx.x;
    const int wave  = tid >> 5;
    const int lane  = tid & 31;
    const int nlane = lane & 15;
    const int row0  = blockIdx.x * 64;

    for (int i = tid; i < (HDIM / 2) * HDIM; i += 128) {
        int p = i >> 6, n = i & 63;
        v2f w0, w1;
        w0.x = A0[(2 * p) * HDIM + n]; w0.y = A0[(2 * p + 1) * HDIM + n];
        w1.x = A1[(2 * p) * HDIM + n]; w1.y = A1[(2 * p + 1) * HDIM + n];
        ((v2f*)sA0)[p * HDIM + n] = w0;
        ((v2f*)sA1)[p * HDIM + n] = w1;
    }
    for (int i = tid; i < 64 * HDIM; i += 128) {
        int r = i >> 6;
        int c = i & 63;
        int gr = row0 + r; if (gr >= N) gr = N - 1;
        float c1 = cnt1[gr]; c1 = (c1 > 0.0f) ? c1 : 1.0f;
        float c2 = cnt2[gr]; c2 = (c2 > 0.0f) ? c2 : 1.0f;
        sM1[r * MPITCH + c] = sum1[(size_t)gr * HDIM + c] / c1;
        sM2[r * MPITCH + c] = sum2[(size_t)gr * HDIM + c] / c2;
    }
    __syncthreads();

    const int    kHalf = (lane < 16) ? 0 : 2;
    const float* m1 = sM1 + (wave * 16 + nlane) * MPITCH;
    const float* m2 = sM2 + (wave * 16 + nlane) * MPITCH;
    const v2f*   p0 = (const v2f*)sA0;
    const v2f*   p1 = (const v2f*)sA1;

    v8f acc1[4] = {}, acc2[4] = {};
    for (int k0 = 0; k0 < HDIM; k0 += 4) {
        v2f a1 = *(const v2f*)(m1 + k0 + kHalf);
        v2f a2 = *(const v2f*)(m2 + k0 + kHalf);
        const v2f* ap0 = p0 + (((k0 + kHalf) >> 1) * HDIM + nlane);
        const v2f* ap1 = p1 + (((k0 + kHalf) >> 1) * HDIM + nlane);
#pragma unroll
        for (int t = 0; t < 4; ++t) {
            acc1[t] = __builtin_amdgcn_wmma_f32_16x16x4_f32(
                false, a1, false, ap0[t * 16], (short)0, acc1[t], false, false);
            acc2[t] = __builtin_amdgcn_wmma_f32_16x16x4_f32(
                false, a2, false, ap1[t * 16], (short)0, acc2[t], false, false);
        }
    }

    const int rhalf = (lane < 16) ? 0 : 8;
    const size_t plane = (size_t)N * HDIM;
#pragma unroll
    for (int t = 0; t < 4; ++t) {
        int col = t * 16 + nlane;
#pragma unroll
        for (int r = 0; r < 8; ++r) {
            int row = row0 + wave * 16 + rhalf + r;
            if (row < N) {
                size_t idx = (size_t)row * HDIM + col;
                float ht = h_tar[idx];
                float hm = h_mask[idx];
                float g1 = acc1[t][r];
                float g2 = acc2[t][r];
                // write-once output planes: keep them out of L2 with NT stores
                __builtin_nontemporal_store(elu1(ht + g1), out + idx);
                __builtin_nontemporal_store(elu1(hm + g1), out + plane + idx);
                __builtin_nontemporal_store(elu1(ht + g2), out + 2 * plane + idx);
                __builtin_nontemporal_store(elu1(hm + g2), out + 3 * plane + idx);
            }
        }
    }
}

extern "C" void kernel_launch(void* const* d_in, const int* in_sizes, int n_in,
                              void* d_out, int out_size, void* d_ws, size_t ws_size,
                              hipStream_t stream) {
    const float* feats0    = (const float*)d_in[0];
    const float* feats1    = (const float*)d_in[1];
    const float* feats2    = (const float*)d_in[2];
    const float* mask_feat = (const float*)d_in[3];
    const int*   src1      = (const int*)d_in[4];
    const int*   dst1      = (const int*)d_in[5];
    const int*   src2      = (const int*)d_in[6];
    const int*   dst2      = (const int*)d_in[7];
    const float* W0 = (const float*)d_in[8];
    const float* b0 = (const float*)d_in[9];
    const float* W1 = (const float*)d_in[10];
    const float* b1 = (const float*)d_in[11];
    const float* W2 = (const float*)d_in[12];
    const float* b2 = (const float*)d_in[13];
    const float* A0 = (const float*)d_in[14];
    const float* A1 = (const float*)d_in[15];

    const int N  = in_sizes[0] / DIN;
    const int N1 = in_sizes[1] / DIN;
    const int N2 = in_sizes[2] / DIN;
    const int E  = in_sizes[4];

    // workspace layout (floats): h_mask | h_nei1 | h_nei2 | sum1 | sum2 | cnt1 | cnt2
    float* ws     = (float*)d_ws;
    float* h_mask = ws;
    float* h_nei1 = h_mask + (size_t)N  * HDIM;
    float* h_nei2 = h_nei1 + (size_t)N1 * HDIM;
    float* sum1   = h_nei2 + (size_t)N2 * HDIM;
    float* sum2   = sum1   + (size_t)N  * HDIM;
    float* cnt1   = sum2   + (size_t)N  * HDIM;
    float* cnt2   = cnt1   + (size_t)N;

    float* out   = (float*)d_out;
    float* h_tar = out + (size_t)4 * N * HDIM;   // output plane 4 doubles as h_tar buffer

    const int smem_gemm = (DIN / 2) * HDIM * 8 + 64 * XPITCH * 4;            // 66560 B
    const int smem_fin  = 2 * (HDIM / 2) * HDIM * 8 + 2 * 64 * MPITCH * 4;   // 67584 B
    hipFuncSetAttribute((const void*)k_gemm_elu,
                        hipFuncAttributeMaxDynamicSharedMemorySize, smem_gemm);
    hipFuncSetAttribute((const void*)k_finalize,
                        hipFuncAttributeMaxDynamicSharedMemorySize, smem_fin);

    // zero accumulators (sum1|sum2|cnt1|cnt2 are contiguous)
    size_t zn = 2 * (size_t)N * HDIM + 2 * (size_t)N;
    k_zero<<<2048, 256, 0, stream>>>(sum1, zn);

    dim3 blk(128);
    k_gemm_elu<<<(N  + 63) / 64, blk, smem_gemm, stream>>>(feats0,    W0, b0, h_tar,  N);
    k_gemm_elu<<<(N  + 63) / 64, blk, smem_gemm, stream>>>(mask_feat, W0, b0, h_mask, N);
    k_gemm_elu<<<(N1 + 63) / 64, blk, smem_gemm, stream>>>(feats1,    W1, b1, h_nei1, N1);
    k_gemm_elu<<<(N2 + 63) / 64, blk, smem_gemm, stream>>>(feats2,    W2, b2, h_nei2, N2);

    long long tE = (long long)E * 32;
    int ablocks = (int)((tE + 255) / 256);
    k_edge_agg<<<ablocks, 256, 0, stream>>>(h_nei1, src1, dst1, sum1, cnt1, E);
    k_edge_agg<<<ablocks, 256, 0, stream>>>(h_nei2, src2, dst2, sum2, cnt2, E);

    k_finalize<<<(N + 63) / 64, blk, smem_fin, stream>>>(sum1, cnt1, sum2, cnt2,
                                                         A0, A1, h_tar, h_mask, out, N);
}